// PoolingModule_9405978378522
// MI455X (gfx1250) — compile-verified
//
#include <hip/hip_runtime.h>
#include <cstdint>

#define KH 3
#define KW 3
#define K2 9
#define B_ 8
#define C_ 256
#define H_ 64
#define W_ 64
#define HW_ 4096   // 64*64
#define CG 4       // channels per workgroup (planes staged in LDS)

// One workgroup per (b, 4-channel group). Stage the 4 contiguous 16 KB x-planes
// (64 KB) into LDS via CDNA5 async global->LDS copies; compute the
// channel-invariant bilinear coefficients ONCE per (pixel, tap) and apply them
// to all 4 channels with LDS gathers. Each thread owns pairs of consecutive
// pixels so offset reads / output writes are b64-vectorized, without the
// register pressure of 4-wide quads.
__global__ __launch_bounds__(256) void deform_dw_kernel(
    const float* __restrict__ x,
    const float* __restrict__ offset,
    const float* __restrict__ weight,
    const float* __restrict__ bias,
    float* __restrict__ out)
{
    const int cg = blockIdx.x;        // 0..63
    const int b  = blockIdx.y;        // 0..7
    const int c0 = cg * CG;

    __shared__ float sh[CG * HW_];    // 64 KB: x[b, c0..c0+3, :, :]

    // ---- CDNA5 async copy: 64 KB contiguous global -> LDS (ASYNCcnt) ----
    const float* gbase = x + (((size_t)b * C_ + c0) * HW_);
    const unsigned lds_base = (unsigned)(uintptr_t)&sh[0];
    #pragma unroll
    for (int j = 0; j < (CG * HW_ * 4) / (256 * 16); ++j) {   // 16 iters
        unsigned boff = (threadIdx.x + (unsigned)j * 256u) << 4;  // 16B per lane
        asm volatile("global_load_async_to_lds_b128 %0, %1, %2 offset:0"
                     :
                     : "v"(lds_base + boff), "v"(boff), "s"(gbase)
                     : "memory");
    }

    // Block-uniform per-channel 3x3 weights + bias (scalar loads).
    float wk[CG][K2];
    float bz[CG];
    #pragma unroll
    for (int ci = 0; ci < CG; ++ci) {
        #pragma unroll
        for (int k = 0; k < K2; ++k) wk[ci][k] = weight[(c0 + ci) * K2 + k];
        bz[ci] = bias[c0 + ci];
    }

    asm volatile("s_wait_asynccnt 0x0" ::: "memory");
    __syncthreads();

    const float* offb = offset + (size_t)b * (2 * K2) * HW_;
    float* outp = out + (((size_t)b * C_ + c0) * HW_);

    // 4096 pixels / (256 threads * 2-pixel pairs) = 8 iterations/thread.
    // Pairs are 8B aligned and never straddle a row (2 | 64).
    #pragma unroll 1
    for (int j = 0; j < 8; ++j) {
        const int p0 = 2 * (int)threadIdx.x + j * 512;
        const int h  = p0 >> 6;
        const int w0 = p0 & 63;

        float acc[2][CG];
        #pragma unroll
        for (int q = 0; q < 2; ++q)
            #pragma unroll
            for (int ci = 0; ci < CG; ++ci) acc[q][ci] = bz[ci];

        #pragma unroll
        for (int k = 0; k < K2; ++k) {
            const float2 oy2 = *(const float2*)&offb[(2 * k + 0) * HW_ + p0];
            const float2 ox2 = *(const float2*)&offb[(2 * k + 1) * HW_ + p0];
            #pragma unroll
            for (int q = 0; q < 2; ++q) {
                const float py = (float)(h - 1 + (k / 3)) + (q ? oy2.y : oy2.x);
                const float px = (float)(w0 + q - 1 + (k % 3)) + (q ? ox2.y : ox2.x);
                const float y0f = floorf(py);
                const float x0f = floorf(px);
                const float ly = py - y0f;
                const float lx = px - x0f;
                const int y0 = (int)y0f;
                const int x0 = (int)x0f;
                const int y1 = y0 + 1;
                const int x1 = x0 + 1;
                // Channel-invariant: validity folded into bilinear weights,
                // clamped LDS indices.
                const float vy0 = (y0 >= 0 && y0 < H_) ? 1.0f : 0.0f;
                const float vy1 = (y1 >= 0 && y1 < H_) ? 1.0f : 0.0f;
                const float vx0 = (x0 >= 0 && x0 < W_) ? 1.0f : 0.0f;
                const float vx1 = (x1 >= 0 && x1 < W_) ? 1.0f : 0.0f;
                const int yc0 = min(max(y0, 0), H_ - 1);
                const int yc1 = min(max(y1, 0), H_ - 1);
                const int xc0 = min(max(x0, 0), W_ - 1);
                const int xc1 = min(max(x1, 0), W_ - 1);
                const int i00 = yc0 * W_ + xc0;
                const int i01 = yc0 * W_ + xc1;
                const int i10 = yc1 * W_ + xc0;
                const int i11 = yc1 * W_ + xc1;
                const float w00 = (1.0f - ly) * (1.0f - lx) * vy0 * vx0;
                const float w01 = (1.0f - ly) * lx * vy0 * vx1;
                const float w10 = ly * (1.0f - lx) * vy1 * vx0;
                const float w11 = ly * lx * vy1 * vx1;
                // Apply the shared coefficients to all 4 staged channels.
                #pragma unroll
                for (int ci = 0; ci < CG; ++ci) {
                    const float* pl = &sh[ci * HW_];
                    const float v00 = pl[i00];
                    const float v01 = pl[i01];
                    const float v10 = pl[i10];
                    const float v11 = pl[i11];
                    acc[q][ci] += wk[ci][k] *
                                  (w00 * v00 + w01 * v01 + w10 * v10 + w11 * v11);
                }
            }
        }
        #pragma unroll
        for (int ci = 0; ci < CG; ++ci) {
            float2 r;
            r.x = acc[0][ci];
            r.y = acc[1][ci];
            *(float2*)&outp[ci * HW_ + p0] = r;   // b64 store
        }
    }
}

extern "C" void kernel_launch(void* const* d_in, const int* in_sizes, int n_in,
                              void* d_out, int out_size, void* d_ws, size_t ws_size,
                              hipStream_t stream) {
    (void)in_sizes; (void)n_in; (void)d_ws; (void)ws_size; (void)out_size;
    const float* x      = (const float*)d_in[0];  // (8, 256, 64, 64)
    const float* offset = (const float*)d_in[1];  // (8, 18, 64, 64)
    const float* weight = (const float*)d_in[2];  // (256, 1, 3, 3)
    const float* bias   = (const float*)d_in[3];  // (256,)
    float* out          = (float*)d_out;          // (8, 256, 64, 64)

    dim3 grid(C_ / CG, B_);   // one workgroup per (b, 4-channel group)
    dim3 block(256);          // 8 wave32s
    deform_dw_kernel<<<grid, block, 0, stream>>>(x, offset, weight, bias, out);
}